// HierarchicalLoss_45792941310261
// MI455X (gfx1250) — compile-verified
//
#include <hip/hip_runtime.h>
#include <stdint.h>

// Hierarchical softmax loss, MI455X (gfx1250).
// Memory-bound: only 80 of 584 floats per row are live (first 72 + leaf group of 8).
// Bulk 72-float row segments moved global->LDS by the Tensor Data Mover (one 2D
// strided tile per wave, TENSORcnt-tracked), with TDM LDS padding (1 dword per 8)
// giving row stride 81 dwords -> bank-conflict-free LDS reads (gcd(81,64)=1).

#define M_COLS 584
#define LEAF_START 73
#define ROWS_PER_BLOCK 128
#define THREADS 128
#define ROW_DW 81            // 72 data dwords + 9 pad dwords per row in LDS
#define NBLOCKS 512          // 65536 / 128

typedef unsigned int u32x4 __attribute__((ext_vector_type(4)));
typedef int i32x8 __attribute__((ext_vector_type(8)));
typedef int i32x4 __attribute__((ext_vector_type(4)));

__device__ __forceinline__ float clampp(float p) {
  return fminf(fmaxf(p, 1e-9f), 1.0f - 1e-9f);
}

__global__ __launch_bounds__(THREADS) void hloss_main(
    const float* __restrict__ inputs, const int* __restrict__ targets,
    float* __restrict__ partial) {
  __shared__ float smem[ROWS_PER_BLOCK * ROW_DW + THREADS];
  float* red = smem + ROWS_PER_BLOCK * ROW_DW;

  const int t = threadIdx.x;
  const int row = blockIdx.x * ROWS_PER_BLOCK + t;

  // ---- TDM: each wave DMAs its 32-row x 72-float tile into LDS ----
  const int wv = __builtin_amdgcn_readfirstlane((int)(threadIdx.x >> 5));
  const uint64_t ga = (uint64_t)(uintptr_t)(inputs +
      (size_t)(blockIdx.x * ROWS_PER_BLOCK + wv * 32) * M_COLS);
  const uint32_t la = (uint32_t)(uintptr_t)&smem[wv * 32 * ROW_DW]; // low 32 bits = LDS offset

  u32x4 g0;
  g0[0] = 1u;                                    // count=1 (valid), no gather
  g0[1] = la;                                    // lds_addr (bytes)
  g0[2] = (uint32_t)ga;                          // global_addr[31:0]
  g0[3] = (uint32_t)(ga >> 32) | (2u << 30);     // global_addr[56:32] | type=2

  i32x8 g1;
  g1[0] = (int)((2u << 16) | (1u << 20) | (2u << 22)); // data_size=4B, pad_en, interval=8dw, amount=1dw
  g1[1] = (int)((uint32_t)(M_COLS & 0xFFFF) << 16);    // tensor_dim0[15:0]
  g1[2] = (int)(((uint32_t)(65536 & 0xFFFF) << 16) | ((M_COLS >> 16) & 0xFFFF)); // dim1 lo | dim0 hi
  g1[3] = (int)(((uint32_t)72u << 16) | ((65536 >> 16) & 0xFFFF));               // tile_dim0=72 | dim1 hi
  g1[4] = 32;                                    // tile_dim1=32 rows, tile_dim2=0
  g1[5] = M_COLS;                                // tensor_dim0_stride lo32 (elements)
  g1[6] = 0;
  g1[7] = 0;
  i32x4 g2 = {0, 0, 0, 0};
  i32x4 g3 = {0, 0, 0, 0};
#if defined(__clang_major__) && (__clang_major__ >= 23)
  i32x8 g4 = {0, 0, 0, 0, 0, 0, 0, 0};
  __builtin_amdgcn_tensor_load_to_lds(g0, g1, g2, g3, g4, 0);
#else
  __builtin_amdgcn_tensor_load_to_lds(g0, g1, g2, g3, 0);
#endif

  // ---- leaf-group gather (overlaps the TDM DMA) ----
  const int tgt = targets[row];
  const int leaf = tgt + LEAF_START;
  const int a2 = (leaf - 1) >> 3;   // 9..72 : parent of leaf == leaf's group id
  const int a1 = (a2 - 1) >> 3;     // 1..8
  const float4* gl = (const float4*)(inputs + (size_t)row * M_COLS + (a2 << 3));
  const float4 L0 = gl[0];
  const float4 L1 = gl[1];

  __builtin_amdgcn_s_wait_tensorcnt(0);
  __syncthreads();

  const float* lrow = smem + t * ROW_DW;

  // group 0 (nodes 1..8): keep clipped probs (parents of level-2 groups)
  float x[8];
#pragma unroll
  for (int e = 0; e < 8; ++e) x[e] = lrow[e];
  float m = x[0];
#pragma unroll
  for (int e = 1; e < 8; ++e) m = fmaxf(m, x[e]);
  float s = 0.f;
#pragma unroll
  for (int e = 0; e < 8; ++e) { x[e] = __expf(x[e] - m); s += x[e]; }
  const float inv = 1.0f / s;
  float c1[8];
  float reg = 0.f;
#pragma unroll
  for (int e = 0; e < 8; ++e) { c1[e] = clampp(x[e] * inv); reg += c1[e]; }
  float ca1 = c1[0];
#pragma unroll
  for (int e = 1; e < 8; ++e) if ((a1 - 1) == e) ca1 = c1[e];

  // groups 1..8 (nodes 9..72): reg += (sum of clipped probs) * parent prob
  float mA = 0.f, sA = 1.f;
#pragma unroll
  for (int g = 1; g < 9; ++g) {
    float y[8];
#pragma unroll
    for (int e = 0; e < 8; ++e) y[e] = lrow[9 * g + e];
    float mg = y[0];
#pragma unroll
    for (int e = 1; e < 8; ++e) mg = fmaxf(mg, y[e]);
    float sg = 0.f;
#pragma unroll
    for (int e = 0; e < 8; ++e) { y[e] = __expf(y[e] - mg); sg += y[e]; }
    const float invg = 1.0f / sg;
    float acc = 0.f;
#pragma unroll
    for (int e = 0; e < 8; ++e) acc += clampp(y[e] * invg);
    reg += acc * c1[g - 1];
    if (g == a1) { mA = mg; sA = sg; }  // capture a2's group stats
  }
  const int posA2 = (a2 - 1) & 7;
  const float xa2 = lrow[9 * a1 + posA2];          // dynamic LDS read, fine
  const float ca2 = clampp(__expf(xa2 - mA) / sA);

  // leaf group softmax from the gathered 8 floats
  float xl[8] = {L0.x, L0.y, L0.z, L0.w, L1.x, L1.y, L1.z, L1.w};
  float ml = xl[0];
#pragma unroll
  for (int e = 1; e < 8; ++e) ml = fmaxf(ml, xl[e]);
  float sl = 0.f;
#pragma unroll
  for (int e = 0; e < 8; ++e) { xl[e] = __expf(xl[e] - ml); sl += xl[e]; }
  const float invl = 1.0f / sl;
  const int posL = (leaf - 1) & 7;
  float cl = 1.f;
#pragma unroll
  for (int e = 0; e < 8; ++e) { float c = clampp(xl[e] * invl); if (e == posL) cl = c; }

  const float ce = -(3.0f * __logf(ca1) + 2.0f * __logf(ca2) + __logf(cl));
  const float v = ce + reg;

  // block reduction
  red[t] = v;
  __syncthreads();
  for (int off = THREADS / 2; off > 0; off >>= 1) {
    if (t < off) red[t] += red[t + off];
    __syncthreads();
  }
  if (t == 0) partial[blockIdx.x] = red[0];
}

__global__ __launch_bounds__(256) void hloss_finish(const float* __restrict__ partial,
                                                    float* __restrict__ out) {
  __shared__ float red[256];
  const int t = threadIdx.x;
  red[t] = partial[t] + partial[t + 256];   // NBLOCKS == 512
  __syncthreads();
  for (int off = 128; off > 0; off >>= 1) {
    if (t < off) red[t] += red[t + off];
    __syncthreads();
  }
  if (t == 0) out[0] = red[0] * (1.0f / 65536.0f);
}

extern "C" void kernel_launch(void* const* d_in, const int* in_sizes, int n_in,
                              void* d_out, int out_size, void* d_ws, size_t ws_size,
                              hipStream_t stream) {
  (void)in_sizes; (void)n_in; (void)out_size; (void)ws_size;
  const float* inputs = (const float*)d_in[0];
  const int* targets = (const int*)d_in[1];
  float* partial = (float*)d_ws;   // 512 floats, fully overwritten each call
  hloss_main<<<NBLOCKS, THREADS, 0, stream>>>(inputs, targets, partial);
  hloss_finish<<<1, 256, 0, stream>>>(partial, (float*)d_out);
}